// Detector_91001767067966
// MI455X (gfx1250) — compile-verified
//
#include <hip/hip_runtime.h>
#include <hip/hip_bf16.h>
#include <math.h>

typedef __attribute__((ext_vector_type(2))) float v2f;
typedef __attribute__((ext_vector_type(8))) float v8f;

#define NUM_CLASS 80
#define NLEV 5
#define KTOT 4256
#define DET_NMS_TH 0.05f
#define DET_NMS_IOU 0.5f
#define DET_NMS_ITERS 15
#define NUMDETS 100
#define CHUNK 512

__device__ __forceinline__ unsigned sortable_key(float f){
  unsigned b = __float_as_uint(f);
  return (b & 0x80000000u) ? ~b : (b | 0x80000000u);
}

// ---------- kernel 1: per-anchor class argmax (logits) + sigmoid ----------
__global__ void score_kernel(const float* __restrict__ cls,
                             float* __restrict__ ps, int* __restrict__ pi, int n){
  int i = blockIdx.x*blockDim.x + threadIdx.x;
  if (i >= n) return;
  const float4* p = reinterpret_cast<const float4*>(cls) + (size_t)i*(NUM_CLASS/4);
  float best = -3.0e38f; int bi = 0;
#pragma unroll
  for (int c = 0; c < NUM_CLASS/4; ++c){
    float4 v = p[c];
    if (v.x > best){ best = v.x; bi = 4*c+0; }
    if (v.y > best){ best = v.y; bi = 4*c+1; }
    if (v.z > best){ best = v.z; bi = 4*c+2; }
    if (v.w > best){ best = v.w; bi = 4*c+3; }
  }
  ps[i] = 1.0f/(1.0f + expf(-best));
  pi[i] = bi + 1;
}

// ---------- kernel 2: exact per-level top-k via radix select ----------
__global__ __launch_bounds__(256) void topk_kernel(
    const float* __restrict__ ps_all, const int* __restrict__ pi_all,
    const float* __restrict__ r0, const float* __restrict__ r1,
    const float* __restrict__ r2, const float* __restrict__ r3,
    const float* __restrict__ r4, const float* __restrict__ loc,
    float* __restrict__ cand_score, int* __restrict__ cand_class,
    int* __restrict__ cand_valid, float4* __restrict__ cand_box){
  const int ns[NLEV]   = {65536,16384,4096,1024,256};
  const int aoff[NLEV] = {0,65536,81920,86016,87040};
  const int kk[NLEV]   = {1000,1000,1000,1000,256};
  const int ooff[NLEV] = {0,1000,2000,3000,4000};
  int lvl = blockIdx.x;
  int n = ns[lvl], k = kk[lvl], ai = aoff[lvl], oo = ooff[lvl];
  const float* reg = (lvl==0)?r0:(lvl==1)?r1:(lvl==2)?r2:(lvl==3)?r3:r4;
  float vy1=loc[0], vx1=loc[1], vy2=loc[2], vx2=loc[3];

  __shared__ unsigned hist[256];
  __shared__ unsigned s_prefix;
  __shared__ int s_remaining;
  __shared__ unsigned s_above, s_eq;
  int tid = threadIdx.x, bs = blockDim.x;
  if (tid==0){ s_prefix=0u; s_remaining=k; s_above=0u; s_eq=0u; }
  __syncthreads();
  for (int pass=3; pass>=0; --pass){
    int shift = pass*8;
    for (int b=tid;b<256;b+=bs) hist[b]=0u;
    __syncthreads();
    unsigned pfx = s_prefix;
    unsigned himask = (pass==3) ? 0u : (0xFFFFFFFFu << (shift+8));
    for (int i=tid;i<n;i+=bs){
      float p = ps_all[ai+i];
      float f = (p > DET_NMS_TH) ? p : -1.0f;
      unsigned key = sortable_key(f);
      if ((key & himask) == (pfx & himask))
        atomicAdd(&hist[(key>>shift)&255u], 1u);
    }
    __syncthreads();
    if (tid==0){
      int rem = s_remaining, cum = 0;
      for (int b=255;b>=0;--b){
        int c = (int)hist[b];
        if (cum + c >= rem){ s_prefix = pfx | ((unsigned)b << shift); s_remaining = rem - cum; break; }
        cum += c;
      }
    }
    __syncthreads();
  }
  unsigned T = s_prefix;
  int need_eq = s_remaining;
  int n_above = k - need_eq;
  for (int i=tid;i<n;i+=bs){
    float p = ps_all[ai+i];
    bool valid = (p > DET_NMS_TH);
    float f = valid ? p : -1.0f;
    unsigned key = sortable_key(f);
    int slot = -1;
    if (key > T) slot = (int)atomicAdd(&s_above, 1u);
    else if (key == T){
      int e = (int)atomicAdd(&s_eq, 1u);
      if (e < need_eq) slot = n_above + e;
    }
    if (slot >= 0){
      int o = oo + slot;
      cand_score[o] = valid ? p : 0.0f;   // where(valid, score, 0)
      cand_class[o] = pi_all[ai+i];
      cand_valid[o] = valid ? 1 : 0;
      const float* b = reg + (size_t)i*4;
      float y1 = fmaxf(b[0], vy1);        // clip to valid window
      float x1 = fmaxf(b[1], vx1);
      float y2 = fminf(b[2], vy2);
      float x2 = fminf(b[3], vx2);
      cand_box[o] = make_float4(y1, x1, y2, x2);
    }
  }
}

// ---------- kernel 3: global bitonic sort (score desc, idx asc) ----------
__global__ __launch_bounds__(1024) void sort_kernel(
    const float* __restrict__ cand_score, const int* __restrict__ cand_class,
    const int* __restrict__ cand_valid, const float4* __restrict__ cand_box,
    float* __restrict__ s_score, int* __restrict__ s_class, int* __restrict__ s_valid,
    float* __restrict__ s_area, float4* __restrict__ s_box, float* __restrict__ keep0){
  extern __shared__ unsigned long long keys[];
  const int N = 8192;
  int tid = threadIdx.x;
  for (int i=tid;i<N;i+=1024){
    if (i < KTOT){
      unsigned sb = __float_as_uint(cand_score[i]);  // scores >= 0: bit order == numeric order
      keys[i] = ((unsigned long long)sb << 32) | (unsigned long long)(0xFFFFFFFFu - (unsigned)i);
    } else keys[i] = 0ull;                           // pads sort last
  }
  __syncthreads();
  for (unsigned k2=2; k2<=(unsigned)N; k2<<=1){
    for (unsigned j=k2>>1; j>0; j>>=1){
      for (unsigned i=tid;i<(unsigned)N;i+=1024){
        unsigned ixj = i ^ j;
        if (ixj > i){
          unsigned long long a=keys[i], b=keys[ixj];
          bool ddir = ((i & k2) == 0);
          if ((a < b) == ddir){ keys[i]=b; keys[ixj]=a; }
        }
      }
      __syncthreads();
    }
  }
  for (int i=tid;i<KTOT;i+=1024){
    unsigned long long kv = keys[i];
    unsigned idx = 0xFFFFFFFFu - (unsigned)(kv & 0xFFFFFFFFull);
    s_score[i] = __uint_as_float((unsigned)(kv >> 32));
    s_class[i] = cand_class[idx];
    s_valid[i] = cand_valid[idx];
    float4 b = cand_box[idx];
    s_box[i] = b;
    s_area[i] = fmaxf(b.z-b.x,0.f)*fmaxf(b.w-b.y,0.f);
    keep0[i] = 1.0f;                                 // iteration 1 sees all rows
  }
}

// ---------- kernel 4: one Cluster-NMS column-max round ----------
// Division-free: iou > 0.5  <=>  inter > 0.5*max(union,1e-9).
// keep[i]==0 rows are staged as degenerate (0,0,0,0) boxes (boxes are
// clipped to [0,2047], so inter==0 -> t = -0.5*uniC < 0 -> never suppress),
// removing the per-pair keep load/multiply.
// The triangle mask is hoisted: tiles strictly below the diagonal need no
// row-index compare at all; only the single diagonal tile per wave uses a
// lane-local constant mask.
// Per 16x16 (i,j) tile one V_WMMA_F32_16X16X4_F32 computes the outer-sum
// area_i + area_j (A=[area_i,1,0,0], B=[1;area_j;0;0]); its D layout
// matches the per-lane rows evaluated with VALU min/max.
__global__ __launch_bounds__(128) void nms_iter_kernel(
    const float4* __restrict__ sbox, const float* __restrict__ sarea,
    const float* __restrict__ keep_in, float* __restrict__ keep_out){
  __shared__ float4 cb[CHUNK];
  __shared__ float  ca[CHUNK];
  int tid  = threadIdx.x;
  int lane = tid & 31;
  int wave = tid >> 5;
  int jt = ((int)blockIdx.x*4 + wave)*16;
  bool wactive = (jt < KTOT);                 // wave-uniform: EXEC stays all-1 inside
  int j = jt + (lane & 15);
  float4 bj = make_float4(0.f,0.f,0.f,0.f);
  float areaj = 0.f;
  if (wactive){ bj = sbox[j]; areaj = sarea[j]; }
  float m = -1.0f;                            // max of (inter - 0.5*uniC) over masked pairs
  float lmask = (lane < 16) ? 1.0f : 0.0f;    // zero K=2,3 lanes of A/B
  int off8 = (lane >> 4) << 3;                // D rows: M = v + 8*(lane>=16)
  int lcol = lane & 15;                       // column within tile
  int rows_end = ((int)blockIdx.x*4 + 4)*16;  // rows needed by any wave in block
  if (rows_end > KTOT) rows_end = KTOT;
  v2f B;                                      // loop-invariant B operand
  B.x = lmask;
  B.y = lmask * areaj;
  for (int base=0; base<rows_end; base+=CHUNK){
    int cnt = KTOT - base; if (cnt > CHUNK) cnt = CHUNK;
    for (int t=tid;t<cnt;t+=128){
      float kp = keep_in[base+t];
      float4 b = sbox[base+t];
      cb[t] = (kp > 0.5f) ? b : make_float4(0.f,0.f,0.f,0.f);
      ca[t] = sarea[base+t];
    }
    __syncthreads();
    if (wactive){
      int diag = jt - base;                   // chunk offset of the diagonal tile
      int full_end = (diag < cnt) ? diag : cnt;   // tiles fully below diagonal
      // ---- full tiles: mask is constant-true, no index math ----
      for (int i0=0; i0<full_end; i0+=16){
        v8f U;
#if defined(__gfx1250__) && __has_builtin(__builtin_amdgcn_wmma_f32_16x16x4_f32)
        v2f A;
        A.x = lmask * ca[i0 + lcol];          // A[M][0] = area_i[M]
        A.y = lmask;                          // A[M][1] = 1
        v8f C = {0.f,0.f,0.f,0.f,0.f,0.f,0.f,0.f};
        U = __builtin_amdgcn_wmma_f32_16x16x4_f32(false, A, false, B, (short)0, C, false, false);
#else
#pragma unroll
        for (int v=0; v<8; ++v) U[v] = ca[i0 + off8 + v] + areaj;
#endif
#pragma unroll
        for (int v=0; v<8; ++v){
          float4 bi = cb[i0 + off8 + v];
          float iy1 = fmaxf(bi.x, bj.x);
          float ix1 = fmaxf(bi.y, bj.y);
          float iy2 = fminf(bi.z, bj.z);
          float ix2 = fminf(bi.w, bj.w);
          float inter = fmaxf(iy2-iy1, 0.f) * fmaxf(ix2-ix1, 0.f);
          float uniC = fmaxf(U[v] - inter, 1e-9f);
          m = fmaxf(m, inter - 0.5f*uniC);    // >0  <=>  iou > 0.5 (suppress)
        }
      }
      // ---- diagonal tile: lane-local constant mask (off8+v) < lcol ----
      if (diag >= 0 && diag < cnt){
        int i0 = diag;
        v8f U;
#if defined(__gfx1250__) && __has_builtin(__builtin_amdgcn_wmma_f32_16x16x4_f32)
        v2f A;
        A.x = lmask * ca[i0 + lcol];
        A.y = lmask;
        v8f C = {0.f,0.f,0.f,0.f,0.f,0.f,0.f,0.f};
        U = __builtin_amdgcn_wmma_f32_16x16x4_f32(false, A, false, B, (short)0, C, false, false);
#else
#pragma unroll
        for (int v=0; v<8; ++v) U[v] = ca[i0 + off8 + v] + areaj;
#endif
#pragma unroll
        for (int v=0; v<8; ++v){
          float4 bi = cb[i0 + off8 + v];
          float iy1 = fmaxf(bi.x, bj.x);
          float ix1 = fmaxf(bi.y, bj.y);
          float iy2 = fminf(bi.z, bj.z);
          float ix2 = fminf(bi.w, bj.w);
          float inter = fmaxf(iy2-iy1, 0.f) * fmaxf(ix2-ix1, 0.f);
          float uniC = fmaxf(U[v] - inter, 1e-9f);
          float t = inter - 0.5f*uniC;
          float tm = ((off8 + v) < lcol) ? t : -1.0f;  // strict upper triangle
          m = fmaxf(m, tm);
        }
      }
    }
    __syncthreads();
  }
  if (wactive){
    m = fmaxf(m, __shfl_xor(m, 16, 32));      // merge row-halves per column
    if (lane < 16) keep_out[j] = (m > 0.0f) ? 0.0f : 1.0f;
  }
}

// ---------- kernel 5: top-100 selection + rescale ----------
__global__ __launch_bounds__(128) void finalize_kernel(
    const float* __restrict__ s_score, const int* __restrict__ s_class,
    const int* __restrict__ s_valid, const float4* __restrict__ s_box,
    const float* __restrict__ keep, const float* __restrict__ loc,
    float* __restrict__ out){
  __shared__ int sel[NUMDETS];
  __shared__ int skept[NUMDETS];
  if (threadIdx.x == 0){
    int cnt = 0;
    for (int i=0;i<KTOT && cnt<NUMDETS;++i)
      if (keep[i] > 0.5f && s_valid[i]){ sel[cnt]=i; skept[cnt]=1; ++cnt; }
    for (int i=0;i<KTOT && cnt<NUMDETS;++i)      // zero-score fills, index order
      if (!(keep[i] > 0.5f && s_valid[i])){ sel[cnt]=i; skept[cnt]=0; ++cnt; }
  }
  __syncthreads();
  int t = threadIdx.x;
  if (t < NUMDETS){
    float vy1=loc[0], vx1=loc[1], vy2=loc[2], vx2=loc[3], oh=loc[4], ow=loc[5];
    float sy = oh / fmaxf(vy2 - vy1, 1e-6f);
    float sx = ow / fmaxf(vx2 - vx1, 1e-6f);
    int i = sel[t];
    float4 b = s_box[i];
    out[t*4+0] = (b.x - vy1)*sy;
    out[t*4+1] = (b.y - vx1)*sx;
    out[t*4+2] = (b.z - vy1)*sy;
    out[t*4+3] = (b.w - vx1)*sx;
    out[400+t] = (float)s_class[i];
    out[500+t] = skept[t] ? s_score[i] : 0.0f;
  }
}

extern "C" void kernel_launch(void* const* d_in, const int* in_sizes, int n_in,
                              void* d_out, int out_size, void* d_ws, size_t ws_size,
                              hipStream_t stream){
  // setup_inputs() dict order is interleaved: cls0, reg0, cls1, reg1, ..., locations
  const float* cls[5]; const float* reg[5];
  for (int s=0;s<5;++s){ cls[s] = (const float*)d_in[2*s]; reg[s] = (const float*)d_in[2*s+1]; }
  const float* loc = (const float*)d_in[10];

  const int ns[5]   = {65536,16384,4096,1024,256};
  const int aoff[5] = {0,65536,81920,86016,87040};
  const int NANCH = 87296;

  char* w = (char*)d_ws;
  size_t o = 0;
  auto alloc = [&](size_t bytes)->void*{ o = (o + 255) & ~(size_t)255; void* p = (void*)(w + o); o += bytes; return p; };
  float*  ps_all     = (float*) alloc((size_t)NANCH*sizeof(float));
  int*    pi_all     = (int*)   alloc((size_t)NANCH*sizeof(int));
  float*  cand_score = (float*) alloc(KTOT*sizeof(float));
  int*    cand_class = (int*)   alloc(KTOT*sizeof(int));
  int*    cand_valid = (int*)   alloc(KTOT*sizeof(int));
  float4* cand_box   = (float4*)alloc(KTOT*sizeof(float4));
  float*  s_score    = (float*) alloc(KTOT*sizeof(float));
  int*    s_class    = (int*)   alloc(KTOT*sizeof(int));
  int*    s_valid    = (int*)   alloc(KTOT*sizeof(int));
  float*  s_area     = (float*) alloc(KTOT*sizeof(float));
  float4* s_box      = (float4*)alloc(KTOT*sizeof(float4));
  float*  keepA      = (float*) alloc(KTOT*sizeof(float));
  float*  keepB      = (float*) alloc(KTOT*sizeof(float));

  for (int s=0;s<5;++s){
    int n = ns[s];
    score_kernel<<<(n+255)/256, 256, 0, stream>>>(cls[s], ps_all + aoff[s], pi_all + aoff[s], n);
  }
  topk_kernel<<<5, 256, 0, stream>>>(ps_all, pi_all, reg[0],reg[1],reg[2],reg[3],reg[4], loc,
                                     cand_score, cand_class, cand_valid, cand_box);
  sort_kernel<<<1, 1024, 8192*sizeof(unsigned long long), stream>>>(
      cand_score, cand_class, cand_valid, cand_box,
      s_score, s_class, s_valid, s_area, s_box, keepA);

  // 15 scan iterations + the final B.max(axis=0) outside the scan = 16 rounds
  float* kin = keepA; float* kout = keepB;
  const int nblocks = (KTOT/16 + 3) / 4;
  for (int it = 0; it < DET_NMS_ITERS + 1; ++it){
    nms_iter_kernel<<<nblocks, 128, 0, stream>>>(s_box, s_area, kin, kout);
    float* t = kin; kin = kout; kout = t;
  }
  finalize_kernel<<<1, 128, 0, stream>>>(s_score, s_class, s_valid, s_box, kin, loc, (float*)d_out);
}